// VisionAttention_8126078124505
// MI455X (gfx1250) — compile-verified
//
#include <hip/hip_runtime.h>
#include <math.h>

// ---------------------------------------------------------------------------
// VisionAttention for MI455X (gfx1250, wave32).
// All matmuls use native fp32 WMMA: V_WMMA_F32_16X16X4_F32 (full fp32 path).
//   A frag (16x4 f32):  2 VGPRs. lane<16: M=lane, K={0,1}; lane>=16: M=lane-16, K={2,3}
//   B frag (4x16 f32):  2 VGPRs. lane<16: N=lane, K rows {0,1}; lane>=16: K rows {2,3}
//   C/D   (16x16 f32):  8 VGPRs. row M = r + 8*(lane>>4), col N = lane&15
// Staging is float4 (global_load_b128 / ds_store_b128) and software-pipelined:
// next tile is prefetched into registers while WMMAs consume the current tile.
// ---------------------------------------------------------------------------

typedef __attribute__((ext_vector_type(2))) float v2f;
typedef __attribute__((ext_vector_type(8))) float v8f;

#define WMMA_F32(a, b, c)                                                      \
  __builtin_amdgcn_wmma_f32_16x16x4_f32(false, (a), false, (b), (short)0, (c), \
                                        false, false)

namespace cfg {
constexpr int B = 8, N = 1024, C = 1024, H = 16, D = 64;
constexpr float EPS = 1e-6f;
constexpr float SCALE = 0.125f;  // D^-0.5
}  // namespace cfg

// ---------------------------------------------------------------------------
// Tiled GEMM:  Cmat[M x Nc] = A[M x K] @ Bm[K x Nc] + bias[Nc]
// Block = 256 threads (8 waves), tile 128(M) x 64(N), BK = 16.
// Each wave owns a 32x32 sub-tile -> 2x2 WMMA 16x16 accumulators.
// ---------------------------------------------------------------------------
__global__ __launch_bounds__(256) void gemm128x64(const float* __restrict__ A,
                                                  const float* __restrict__ Bm,
                                                  const float* __restrict__ bias,
                                                  float* __restrict__ Cmat,
                                                  int K, int Nc) {
  // Row pads keep float4 stores 16B-aligned and frag column reads
  // bank-conflict-free (20*L mod 64 and 68*L mod 64 distinct for L=0..15).
  __shared__ __align__(16) float As[128][20];
  __shared__ __align__(16) float Bs[16][68];

  const int tid = threadIdx.x;
  const int lane = tid & 31;
  const int wave = tid >> 5;
  const int hi = lane >> 4;
  const int l16 = lane & 15;
  const int wm = (wave >> 1) * 32;  // 0,32,64,96
  const int wn = (wave & 1) * 32;   // 0,32
  const int m0 = blockIdx.y * 128;
  const int n0 = blockIdx.x * 64;

  // Staging assignment: A tile 128x16 = 512 float4 (2/thread),
  //                     B tile  16x64 = 256 float4 (1/thread).
  const int ar = tid >> 2;             // A rows ar and ar+64
  const int ac = (tid & 3) * 4;
  const int bk = tid >> 4;
  const int bc = (tid & 15) * 4;
  const float* aptr0 = A + (size_t)(m0 + ar) * K + ac;
  const float* aptr1 = A + (size_t)(m0 + ar + 64) * K + ac;
  const float* bptr = Bm + (size_t)bk * Nc + n0 + bc;

  float4 a0 = *(const float4*)aptr0;
  float4 a1 = *(const float4*)aptr1;
  float4 b0 = *(const float4*)bptr;

  v8f acc[2][2] = {};
  const int nk = K / 16;
  for (int t = 0; t < nk; ++t) {
    *(float4*)&As[ar][ac] = a0;
    *(float4*)&As[ar + 64][ac] = a1;
    *(float4*)&Bs[bk][bc] = b0;
    __syncthreads();

    // Prefetch next tile into registers; waits land at next iteration's
    // LDS stores, overlapping global latency with the WMMA work below.
    if (t + 1 < nk) {
      aptr0 += 16;
      aptr1 += 16;
      bptr += (size_t)16 * Nc;
      a0 = *(const float4*)aptr0;
      a1 = *(const float4*)aptr1;
      b0 = *(const float4*)bptr;
    }

#pragma unroll
    for (int ks = 0; ks < 4; ++ks) {
      const int kf = ks * 4 + 2 * hi;
      v2f fa0, fa1, fb0, fb1;
      fa0.x = As[wm + l16][kf];
      fa0.y = As[wm + l16][kf + 1];
      fa1.x = As[wm + 16 + l16][kf];
      fa1.y = As[wm + 16 + l16][kf + 1];
      fb0.x = Bs[kf][wn + l16];
      fb0.y = Bs[kf + 1][wn + l16];
      fb1.x = Bs[kf][wn + 16 + l16];
      fb1.y = Bs[kf + 1][wn + 16 + l16];
      acc[0][0] = WMMA_F32(fa0, fb0, acc[0][0]);
      acc[0][1] = WMMA_F32(fa0, fb1, acc[0][1]);
      acc[1][0] = WMMA_F32(fa1, fb0, acc[1][0]);
      acc[1][1] = WMMA_F32(fa1, fb1, acc[1][1]);
    }
    __syncthreads();
  }

  // Epilogue: fused bias add; each store = two 64B row segments per wave.
#pragma unroll
  for (int i = 0; i < 2; ++i) {
#pragma unroll
    for (int j = 0; j < 2; ++j) {
      const int nc = n0 + wn + j * 16 + l16;
      const float bv = bias[nc];
#pragma unroll
      for (int r = 0; r < 8; ++r) {
        const int mr = m0 + wm + i * 16 + r + 8 * hi;
        Cmat[(size_t)mr * Nc + nc] = acc[i][j][r] + bv;
      }
    }
  }
}

// ---------------------------------------------------------------------------
// In-place rmsnorm of q (cols [0,C)) and k (cols [C,2C)) of the qkv buffer.
// One block per token row; q additionally folds in SCALE = D^-0.5.
// ---------------------------------------------------------------------------
__global__ __launch_bounds__(256) void qk_rmsnorm(float* __restrict__ qkv,
                                                  const float* __restrict__ qw,
                                                  const float* __restrict__ kw) {
  __shared__ float red[256];
  const int tid = threadIdx.x;
  float* row = qkv + (size_t)blockIdx.x * (3 * cfg::C);
  float4* q4 = (float4*)row;
  float4* k4 = (float4*)(row + cfg::C);
  const float4* qw4 = (const float4*)qw;
  const float4* kw4 = (const float4*)kw;

  // ---- q ----
  float4 v = q4[tid];
  float ss = v.x * v.x + v.y * v.y + v.z * v.z + v.w * v.w;
  red[tid] = ss;
  __syncthreads();
  for (int s = 128; s > 0; s >>= 1) {
    if (tid < s) red[tid] += red[tid + s];
    __syncthreads();
  }
  const float invq = rsqrtf(red[0] * (1.0f / cfg::C) + cfg::EPS) * cfg::SCALE;
  __syncthreads();
  {
    const float4 w = qw4[tid];
    v.x *= w.x * invq;
    v.y *= w.y * invq;
    v.z *= w.z * invq;
    v.w *= w.w * invq;
    q4[tid] = v;
  }

  // ---- k ----
  float4 u = k4[tid];
  ss = u.x * u.x + u.y * u.y + u.z * u.z + u.w * u.w;
  red[tid] = ss;
  __syncthreads();
  for (int s = 128; s > 0; s >>= 1) {
    if (tid < s) red[tid] += red[tid + s];
    __syncthreads();
  }
  const float invk = rsqrtf(red[0] * (1.0f / cfg::C) + cfg::EPS);
  __syncthreads();
  {
    const float4 w = kw4[tid];
    u.x *= w.x * invk;
    u.y *= w.y * invk;
    u.z *= w.z * invk;
    u.w *= w.w * invk;
    k4[tid] = u;
  }
}

// ---------------------------------------------------------------------------
// Flash attention, fp32 WMMA throughout.
// grid = (N/64, B*H); block = 256 threads (8 waves) owns a 64-row q tile.
// Q strip lives in registers (16 K-pair frags per lane); K and V stream
// through separate LDS buffers with register prefetch of the next tile.
// Output written to [B, N, C] so the projection GEMM reads it directly.
// ---------------------------------------------------------------------------
__global__ __launch_bounds__(256) void attention(const float* __restrict__ qkv,
                                                 float* __restrict__ attn_out) {
  __shared__ __align__(16) float Ks[64][68];
  __shared__ __align__(16) float Vs[64][68];
  __shared__ float Ss[64][67];  // 67: 3*L mod 64 distinct, halves disjoint
  __shared__ float rowm[64], rowl[64], rowc[64];

  const int tid = threadIdx.x;
  const int lane = tid & 31;
  const int wave = tid >> 5;
  const int hi = lane >> 4;
  const int l16 = lane & 15;
  const int mt = wave >> 1;        // wave's 16-row output strip (0..3)
  const int ntb = (wave & 1) * 2;  // wave's pair of 16-col output tiles

  const int q0 = blockIdx.x * 64;
  const int bh = blockIdx.y;
  const int b = bh / cfg::H, h = bh % cfg::H;
  const size_t rs = 3 * cfg::C;  // qkv row stride
  const float* qg = qkv + (size_t)b * cfg::N * rs + (size_t)h * cfg::D;
  const float* kg = qg + cfg::C;
  const float* vg = qg + 2 * cfg::C;

  // ---- Q strip -> registers: lane owns row mt*16+l16, K-pairs at 4k+2*hi ----
  v2f qa[16];
  {
    const float* qrow = qg + (size_t)(q0 + mt * 16 + l16) * rs + 2 * hi;
#pragma unroll
    for (int ks = 0; ks < 16; ++ks) {
      const float2 f = *(const float2*)(qrow + ks * 4);  // 8B aligned
      qa[ks].x = f.x;
      qa[ks].y = f.y;
    }
  }

  // ---- Stage K(0), V(0): 64x64 = 1024 float4 each, 4/thread ----
  const int sn = tid >> 4;             // staged row (+16 per step)
  const int sc = (tid & 15) * 4;       // staged float4 column
  float4 kreg[4], vreg[4];
#pragma unroll
  for (int i = 0; i < 4; ++i) {
    kreg[i] = *(const float4*)(kg + (size_t)(sn + 16 * i) * rs + sc);
    vreg[i] = *(const float4*)(vg + (size_t)(sn + 16 * i) * rs + sc);
  }
#pragma unroll
  for (int i = 0; i < 4; ++i) {
    *(float4*)&Ks[sn + 16 * i][sc] = kreg[i];
    *(float4*)&Vs[sn + 16 * i][sc] = vreg[i];
  }
  if (tid < 64) {
    rowm[tid] = -__builtin_inff();
    rowl[tid] = 0.f;
  }
  v8f acc[2] = {};
  __syncthreads();

  for (int kb = 0; kb < cfg::N; kb += 64) {
    const bool more = (kb + 64) < cfg::N;

    // Prefetch next K tile into registers during S compute.
    if (more) {
#pragma unroll
      for (int i = 0; i < 4; ++i)
        kreg[i] = *(const float4*)(kg + (size_t)(kb + 64 + sn + 16 * i) * rs + sc);
    }

    // ---- S = Q @ K^T (each wave: 2 tiles, 16 K-steps of 4) ----
#pragma unroll
    for (int j = 0; j < 2; ++j) {
      const int nt = ntb + j;
      v8f s = {};
#pragma unroll
      for (int ks = 0; ks < 16; ++ks) {
        const float2 f = *(const float2*)&Ks[nt * 16 + l16][ks * 4 + 2 * hi];
        v2f bb;
        bb.x = f.x;
        bb.y = f.y;
        s = WMMA_F32(qa[ks], bb, s);
      }
#pragma unroll
      for (int r = 0; r < 8; ++r)
        Ss[mt * 16 + r + 8 * hi][nt * 16 + l16] = s[r];
    }
    __syncthreads();  // Ss visible; Ks consumed

    // Store next K tile; softmax runs concurrently on 64 threads.
    if (more) {
#pragma unroll
      for (int i = 0; i < 4; ++i) *(float4*)&Ks[sn + 16 * i][sc] = kreg[i];
    }
    if (tid < 64) {
      const float mprev = rowm[tid];
      float mnew = mprev;
#pragma unroll 8
      for (int c = 0; c < 64; ++c) mnew = fmaxf(mnew, Ss[tid][c]);
      const float corr = __expf(mprev - mnew);
      float sum = 0.f;
#pragma unroll 8
      for (int c = 0; c < 64; ++c) {
        const float p = __expf(Ss[tid][c] - mnew);
        Ss[tid][c] = p;
        sum += p;
      }
      rowl[tid] = rowl[tid] * corr + sum;
      rowm[tid] = mnew;
      rowc[tid] = corr;
    }
    // Prefetch next V tile into registers.
    if (more) {
#pragma unroll
      for (int i = 0; i < 4; ++i)
        vreg[i] = *(const float4*)(vg + (size_t)(kb + 64 + sn + 16 * i) * rs + sc);
    }
    __syncthreads();  // P and rowc ready

    // ---- rescale running accumulators, then acc += P @ V ----
#pragma unroll
    for (int j = 0; j < 2; ++j)
#pragma unroll
      for (int r = 0; r < 8; ++r) acc[j][r] *= rowc[mt * 16 + r + 8 * hi];

#pragma unroll
    for (int j = 0; j < 2; ++j) {
      const int nt = ntb + j;
#pragma unroll
      for (int ks = 0; ks < 16; ++ks) {
        const int kf = ks * 4 + 2 * hi;
        v2f a, bb;
        a.x = Ss[mt * 16 + l16][kf];
        a.y = Ss[mt * 16 + l16][kf + 1];
        bb.x = Vs[kf][nt * 16 + l16];
        bb.y = Vs[kf + 1][nt * 16 + l16];
        acc[j] = WMMA_F32(a, bb, acc[j]);
      }
    }
    __syncthreads();  // Vs consumed

    if (more) {
#pragma unroll
      for (int i = 0; i < 4; ++i) *(float4*)&Vs[sn + 16 * i][sc] = vreg[i];
    }
  }

  // ---- normalize by l and store into [B, N, C] layout ----
#pragma unroll
  for (int j = 0; j < 2; ++j) {
    const int nt = ntb + j;
#pragma unroll
    for (int r = 0; r < 8; ++r) {
      const int mrow = mt * 16 + r + 8 * hi;
      const float o = acc[j][r] / rowl[mrow];
      attn_out[((size_t)b * cfg::N + q0 + mrow) * cfg::C + h * cfg::D +
               nt * 16 + l16] = o;
    }
  }
}

// ---------------------------------------------------------------------------
extern "C" void kernel_launch(void* const* d_in, const int* in_sizes, int n_in,
                              void* d_out, int out_size, void* d_ws,
                              size_t ws_size, hipStream_t stream) {
  const float* x      = (const float*)d_in[0];  // [B,N,C]
  const float* W_qkv  = (const float*)d_in[1];  // [C,3C]
  const float* b_qkv  = (const float*)d_in[2];  // [3C]
  const float* q_w    = (const float*)d_in[3];  // [C]
  const float* k_w    = (const float*)d_in[4];  // [C]
  const float* W_proj = (const float*)d_in[5];  // [C,C]
  const float* b_proj = (const float*)d_in[6];  // [C]
  float* out = (float*)d_out;                   // [B,N,C] fp32

  const int M = cfg::B * cfg::N;  // 8192
  float* qkv = (float*)d_ws;                   // 96 MB: [M, 3C]
  float* attn = qkv + (size_t)M * 3 * cfg::C;  // 32 MB: [M, C]

  // 1) qkv = x @ W_qkv + b_qkv
  gemm128x64<<<dim3(3 * cfg::C / 64, M / 128), 256, 0, stream>>>(
      x, W_qkv, b_qkv, qkv, cfg::C, 3 * cfg::C);

  // 2) rmsnorm(q)*SCALE, rmsnorm(k), in place
  qk_rmsnorm<<<M, 256, 0, stream>>>(qkv, q_w, k_w);

  // 3) flash attention -> attn [M, C]
  attention<<<dim3(cfg::N / 64, cfg::B * cfg::H), 256, 0, stream>>>(qkv, attn);

  // 4) out = attn @ W_proj + b_proj
  gemm128x64<<<dim3(cfg::C / 64, M / 128), 256, 0, stream>>>(
      attn, W_proj, b_proj, out, cfg::C, cfg::C);
}